// IBN_DGCNN2d_65343632441547
// MI455X (gfx1250) — compile-verified
//
#include <hip/hip_runtime.h>

#define NPTS  2048
#define BATCH 2
#define KNN   40

typedef __attribute__((ext_vector_type(2))) float v2f;
typedef __attribute__((ext_vector_type(8))) float v8f;

__device__ __forceinline__ float lrelu(float v, float s) { return v >= 0.f ? v : s * v; }

// ---------------- prefix conv: (B,1,10,N) -> (B,4,N) [rows 2,3 zero-padded for K=4 WMMA] ----------------
__global__ void prefix_conv_k(const float* __restrict__ x, const float* __restrict__ Wc,
                              const float* __restrict__ bc, float* __restrict__ h) {
    int t = blockIdx.x * blockDim.x + threadIdx.x;
    if (t >= BATCH * 4 * NPTS) return;
    int n = t % NPTS;
    int r = (t / NPTS) % 4;
    int b = t / (4 * NPTS);
    if (r >= 2) { h[t] = 0.f; return; }          // zero K-padding rows
    float s = bc[0];
    #pragma unroll
    for (int p = 0; p < 5; ++p) {
        #pragma unroll
        for (int q = 0; q < 3; ++q) {
            int nn = n + q - 1;
            if (nn < 0 || nn >= NPTS) continue;
            s += x[((size_t)b * 10 + r * 5 + p) * NPTS + nn] * Wc[p * 3 + q];
        }
    }
    h[t] = lrelu(s, 0.01f);
}

// ---------------- per-column squared norms of X (C,N) ----------------
__global__ void colsq_k(const float* __restrict__ X, float* __restrict__ xx, int C, int N) {
    int n = blockIdx.x * blockDim.x + threadIdx.x;
    if (n >= N) return;
    float s = 0.f;
    for (int c = 0; c < C; ++c) { float v = X[(size_t)c * N + n]; s += v * v; }
    xx[n] = s;
}

// ---------------- WMMA f32 GEMM: C(MxN) = A(MxK) @ B(KxN), K % 4 == 0, N % 64 == 0 ----------------
// One wave computes a 16x64 tile (4 accumulators, A fragment reused 4x per K-step).
// TRANSA: A(m,k) = A[k*lda+m] (used for Gram X^T X).  EPI: fused neg_d = 2*G - xx[m] - xx[n].
template <int TRANSA, int EPI>
__global__ __launch_bounds__(256) void wmma_gemm_f32(
        const float* __restrict__ A, int lda,
        const float* __restrict__ Bm, int ldb,
        float* __restrict__ Cm, int ldc,
        int M, int Nn, int Kd,
        const float* __restrict__ xxRow, const float* __restrict__ xxCol) {
    int wave = blockIdx.x * (blockDim.x >> 5) + (threadIdx.x >> 5);
    int lane = threadIdx.x & 31;
    int tilesN = Nn >> 6;                          // 64-wide N tiles
    int tilesTotal = (M >> 4) * tilesN;
    if (wave >= tilesTotal) return;                // wave-uniform: EXEC all-1 for WMMA
    int tm = (wave / tilesN) << 4;
    int tn = (wave % tilesN) << 6;
    int r  = lane & 15;
    int kh = (lane >> 4) << 1;                     // lanes 0-15 -> K{0,1}; lanes 16-31 -> K{2,3}
    int mrow = tm + r;

    v8f acc[4];
    #pragma unroll
    for (int t = 0; t < 4; ++t) acc[t] = (v8f){0.f,0.f,0.f,0.f,0.f,0.f,0.f,0.f};

    const float* bp = Bm + tn + r;
    for (int k0 = 0; k0 < Kd; k0 += 4) {
        int ka = k0 + kh;
        v2f a;
        if (TRANSA) {
            a.x = A[(size_t)ka * lda + mrow];
            a.y = A[(size_t)(ka + 1) * lda + mrow];
        } else {
            a = *(const v2f*)&A[(size_t)mrow * lda + ka];   // 8B-aligned: lda%4==0, ka even
        }
        const float* b0 = bp + (size_t)ka * ldb;
        const float* b1 = bp + (size_t)(ka + 1) * ldb;
        v2f bb0 = { b0[0],  b1[0]  };
        v2f bb1 = { b0[16], b1[16] };
        v2f bb2 = { b0[32], b1[32] };
        v2f bb3 = { b0[48], b1[48] };
        acc[0] = __builtin_amdgcn_wmma_f32_16x16x4_f32(false, a, false, bb0, (short)0, acc[0], false, false);
        acc[1] = __builtin_amdgcn_wmma_f32_16x16x4_f32(false, a, false, bb1, (short)0, acc[1], false, false);
        acc[2] = __builtin_amdgcn_wmma_f32_16x16x4_f32(false, a, false, bb2, (short)0, acc[2], false, false);
        acc[3] = __builtin_amdgcn_wmma_f32_16x16x4_f32(false, a, false, bb3, (short)0, acc[3], false, false);
    }

    int cm0 = tm + ((lane < 16) ? 0 : 8);          // D layout: VGPR v -> M = cm0+v, N = ncol
    #pragma unroll
    for (int t = 0; t < 4; ++t) {
        int ncol = tn + t * 16 + r;
        #pragma unroll
        for (int v = 0; v < 8; ++v) {
            float val = acc[t][v];
            if (EPI) val = 2.f * val - xxRow[cm0 + v] - xxCol[ncol];
            Cm[(size_t)(cm0 + v) * ldc + ncol] = val;
        }
    }
}

// ---------------- top-40 per row (descending, ties -> lowest index, like lax.top_k) ----------------
__global__ void topk40_k(const float* __restrict__ D, int* __restrict__ idx) {
    int n = blockIdx.x * blockDim.x + threadIdx.x;
    if (n >= NPTS) return;
    float vals[KNN];
    int   ids[KNN];
    #pragma unroll
    for (int i = 0; i < KNN; ++i) { vals[i] = -3.402823466e38f; ids[i] = 0; }
    const float* row = D + (size_t)n * NPTS;
    for (int m = 0; m < NPTS; ++m) {
        float v = row[m];
        if (v > vals[KNN - 1]) {
            int p = KNN - 1;
            while (p > 0 && vals[p - 1] < v) { vals[p] = vals[p - 1]; ids[p] = ids[p - 1]; --p; }
            vals[p] = v; ids[p] = m;
        }
    }
    int* out = idx + (size_t)n * KNN;
    #pragma unroll
    for (int i = 0; i < KNN; ++i) out[i] = ids[i];
}

// ---------------- restack W (O x 2C) -> [Wn ; Wc - Wn] (2O x Cpad), zero-pad cols [C,Cpad) ----------------
__global__ void prep_w_k(const float* __restrict__ W, float* __restrict__ Ws, int O, int C, int Cpad) {
    int t = blockIdx.x * blockDim.x + threadIdx.x;
    if (t >= O * Cpad) return;
    int o = t / Cpad, c = t % Cpad;
    float wn = (c < C) ? W[(size_t)o * 2 * C + c]     : 0.f;
    float wc = (c < C) ? W[(size_t)o * 2 * C + C + c] : 0.f;
    Ws[(size_t)o * Cpad + c]       = wn;
    Ws[(size_t)(O + o) * Cpad + c] = wc - wn;
}

// ---------------- out[o,n] = lrelu( max_j Y1[o, idx[n,j]] + Y2[o,n] + bias[o], 0.2 ) ----------------
__global__ void edge_max_k(const float* __restrict__ Y, const int* __restrict__ idx,
                           const float* __restrict__ bias, float* __restrict__ out, int O, int N) {
    int t = blockIdx.x * blockDim.x + threadIdx.x;
    if (t >= O * N) return;
    int o = t / N, n = t % N;
    const float* Y1 = Y + (size_t)o * N;
    float base = Y[(size_t)(O + o) * N + n] + bias[o];
    const int* id = idx + (size_t)n * KNN;
    float m = -3.402823466e38f;
    #pragma unroll 8
    for (int j = 0; j < KNN; ++j) m = fmaxf(m, Y1[id[j]]);
    out[(size_t)o * N + n] = lrelu(m + base, 0.2f);
}

// ---------------- g[o] = lrelu( max_n gpre[o,n] + b5[o], 0.2 ) ----------------
__global__ void rowmax_lrelu_k(const float* __restrict__ gpre, const float* __restrict__ b5,
                               float* __restrict__ g, int N) {
    __shared__ float red[256];
    int o = blockIdx.x;
    float m = -3.402823466e38f;
    for (int n = threadIdx.x; n < N; n += blockDim.x) m = fmaxf(m, gpre[(size_t)o * N + n]);
    red[threadIdx.x] = m;
    __syncthreads();
    for (int s = blockDim.x >> 1; s > 0; s >>= 1) {
        if ((int)threadIdx.x < s) red[threadIdx.x] = fmaxf(red[threadIdx.x], red[threadIdx.x + s]);
        __syncthreads();
    }
    if (threadIdx.x == 0) g[o] = lrelu(red[0] + b5[o], 0.2f);
}

// ---------------- FC: out[b,i] = lrelu( g[b]·Wfc[i,:] + bfc[i], 0.2 ) ----------------
__global__ void fc_lrelu_k(const float* __restrict__ g, const float* __restrict__ Wfc,
                           const float* __restrict__ bfc, float* __restrict__ out) {
    int t = blockIdx.x * blockDim.x + threadIdx.x;
    if (t >= BATCH * 16384) return;
    int i = t % 16384, b = t / 16384;
    const float* gr = g + (size_t)b * 1024;
    const float* wr = Wfc + (size_t)i * 1024;
    float s = bfc[i];
    for (int c = 0; c < 1024; ++c) s += gr[c] * wr[c];
    out[t] = lrelu(s, 0.2f);
}

// ---------------- fused nearest-2x upsample + 3x3 conv (pad 1), optional lrelu(0.2) ----------------
__global__ void upconv3x3_k(const float* __restrict__ in, const float* __restrict__ W,
                            const float* __restrict__ bias, float* __restrict__ out,
                            int B, int IC, int OC, int IH, int IW, int act) {
    int OH = IH * 2, OW = IW * 2;
    int t = blockIdx.x * blockDim.x + threadIdx.x;
    int total = B * OC * OH * OW;
    if (t >= total) return;
    int x  = t % OW;
    int y  = (t / OW) % OH;
    int oc = (t / (OW * OH)) % OC;
    int b  = t / (OW * OH * OC);
    float s = bias[oc];
    for (int ic = 0; ic < IC; ++ic) {
        const float* ip = in + ((size_t)b * IC + ic) * IH * IW;
        const float* wp = W + ((size_t)oc * IC + ic) * 9;
        #pragma unroll
        for (int dy = 0; dy < 3; ++dy) {
            int yy = y + dy - 1;
            if (yy < 0 || yy >= OH) continue;
            int iy = yy >> 1;
            #pragma unroll
            for (int dx = 0; dx < 3; ++dx) {
                int xx = x + dx - 1;
                if (xx < 0 || xx >= OW) continue;
                s += ip[(size_t)iy * IW + (xx >> 1)] * wp[dy * 3 + dx];
            }
        }
    }
    out[t] = act ? lrelu(s, 0.2f) : s;
}

extern "C" void kernel_launch(void* const* d_in, const int* in_sizes, int n_in,
                              void* d_out, int out_size, void* d_ws, size_t ws_size,
                              hipStream_t stream) {
    (void)in_sizes; (void)n_in; (void)out_size; (void)ws_size;
    const float* x   = (const float*)d_in[0];
    const float* Wc  = (const float*)d_in[1];
    const float* bc  = (const float*)d_in[2];
    const float* Wl[4] = {(const float*)d_in[3], (const float*)d_in[5],
                          (const float*)d_in[7], (const float*)d_in[9]};
    const float* bl[4] = {(const float*)d_in[4], (const float*)d_in[6],
                          (const float*)d_in[8], (const float*)d_in[10]};
    const float* W5  = (const float*)d_in[11];
    const float* b5  = (const float*)d_in[12];
    const float* Wfc = (const float*)d_in[13];
    const float* bfc = (const float*)d_in[14];
    const float* Wd1 = (const float*)d_in[15];
    const float* bd1 = (const float*)d_in[16];
    const float* Wd2 = (const float*)d_in[17];
    const float* bd2 = (const float*)d_in[18];
    const float* Wd3 = (const float*)d_in[19];
    const float* bd3 = (const float*)d_in[20];

    float* ws = (float*)d_ws;
    size_t off = 0;
    auto take = [&](size_t n) { float* p = ws + off; off += (n + 63) & ~(size_t)63; return p; };
    float* hbuf  = take((size_t)BATCH * 4 * NPTS);        // (B,4,N), rows 2-3 zero (K padding)
    float* catb0 = take((size_t)BATCH * 512 * NPTS);      // x1|x2|x3|x4 stacked -> concat is free
    float* dist  = take((size_t)NPTS * NPTS);             // reused per batch/layer (16 MB)
    int*   idxb  = (int*)take((size_t)NPTS * KNN);
    float* xxb   = take((size_t)NPTS);
    float* wstk  = take((size_t)512 * 128);
    float* ybuf  = take((size_t)512 * NPTS);
    float* gpre  = take((size_t)1024 * NPTS);
    float* gbuf  = take((size_t)BATCH * 1024);
    float* fcout = take((size_t)BATCH * 16384);
    float* c1buf = take((size_t)BATCH * 32 * 32 * 32);
    float* c2buf = take((size_t)BATCH * 16 * 64 * 64);

    auto cdiv = [](int a, int b) { return (a + b - 1) / b; };

    prefix_conv_k<<<cdiv(BATCH * 4 * NPTS, 256), 256, 0, stream>>>(x, Wc, bc, hbuf);

    const int layCr[4]  = {2, 64, 64, 128};               // real input channels
    const int layCe[4]  = {4, 64, 64, 128};               // effective (padded) K
    const int layO[4]   = {64, 64, 128, 256};
    const int inOff[4]  = {-1, 0, 64, 128};
    const int outOff[4] = {0, 64, 128, 256};

    for (int b = 0; b < BATCH; ++b) {
        float* catB = catb0 + (size_t)b * 512 * NPTS;
        for (int L = 0; L < 4; ++L) {
            int Cr = layCr[L], Ce = layCe[L], O = layO[L];
            const float* X = (L == 0) ? (hbuf + (size_t)b * 4 * NPTS)
                                      : (catB + (size_t)inOff[L] * NPTS);
            colsq_k<<<cdiv(NPTS, 256), 256, 0, stream>>>(X, xxb, Ce, NPTS);
            int gwaves = (NPTS / 16) * (NPTS / 64);
            wmma_gemm_f32<1, 1><<<cdiv(gwaves, 8), 256, 0, stream>>>(
                X, NPTS, X, NPTS, dist, NPTS, NPTS, NPTS, Ce, xxb, xxb);
            topk40_k<<<cdiv(NPTS, 128), 128, 0, stream>>>(dist, idxb);
            prep_w_k<<<cdiv(O * Ce, 256), 256, 0, stream>>>(Wl[L], wstk, O, Cr, Ce);
            int ywaves = (2 * O / 16) * (NPTS / 64);
            wmma_gemm_f32<0, 0><<<cdiv(ywaves, 8), 256, 0, stream>>>(
                wstk, Ce, X, NPTS, ybuf, NPTS, 2 * O, NPTS, Ce, nullptr, nullptr);
            edge_max_k<<<cdiv(O * NPTS, 256), 256, 0, stream>>>(
                ybuf, idxb, bl[L], catB + (size_t)outOff[L] * NPTS, O, NPTS);
        }
        int wwaves = (1024 / 16) * (NPTS / 64);
        wmma_gemm_f32<0, 0><<<cdiv(wwaves, 8), 256, 0, stream>>>(
            W5, 512, catB, NPTS, gpre, NPTS, 1024, NPTS, 512, nullptr, nullptr);
        rowmax_lrelu_k<<<1024, 256, 0, stream>>>(gpre, b5, gbuf + (size_t)b * 1024, NPTS);
    }

    fc_lrelu_k<<<cdiv(BATCH * 16384, 256), 256, 0, stream>>>(gbuf, Wfc, bfc, fcout);
    upconv3x3_k<<<cdiv(BATCH * 32 * 32 * 32, 256), 256, 0, stream>>>(
        fcout, Wd1, bd1, c1buf, BATCH, 64, 32, 16, 16, 1);
    upconv3x3_k<<<cdiv(BATCH * 16 * 64 * 64, 256), 256, 0, stream>>>(
        c1buf, Wd2, bd2, c2buf, BATCH, 32, 16, 32, 32, 1);
    upconv3x3_k<<<cdiv(BATCH * 1 * 128 * 128, 256), 256, 0, stream>>>(
        c2buf, Wd3, bd3, (float*)d_out, BATCH, 16, 1, 64, 64, 0);
}